// MixHopLayer_62689342652838
// MI455X (gfx1250) — compile-verified
//
#include <hip/hip_runtime.h>

typedef float v2f __attribute__((ext_vector_type(2)));
typedef float v8f __attribute__((ext_vector_type(8)));

#define C_DIM 128
#define O_DIM 128
#define OUT_STRIDE 384

// ---------------------------------------------------------------------------
// Zero-fill (atomic-accumulation destinations are poisoned by the harness)
// ---------------------------------------------------------------------------
__global__ __launch_bounds__(256) void mixhop_zero(float* __restrict__ p, long n) {
  long i = (long)blockIdx.x * blockDim.x + threadIdx.x;
  long stride = (long)gridDim.x * blockDim.x;
  for (; i < n; i += stride) p[i] = 0.0f;
}

// ---------------------------------------------------------------------------
// Fused triple GEMM: reads x once, produces
//   y0 = x@w0+b0 -> out columns [0,128) of the N x 384 output
//   y1 = x@w1+b1 -> ws
//   y2 = x@w2+b2 -> ws
// One wave computes one 16x16 tile via 32 chained V_WMMA_F32_16X16X4_F32 ops
// per output matrix. 8 waves/block tile the full 128 output columns.
// ---------------------------------------------------------------------------
__global__ __launch_bounds__(256) void mixhop_gemm3(
    const float* __restrict__ x,
    const float* __restrict__ w0, const float* __restrict__ b0,
    const float* __restrict__ w1, const float* __restrict__ b1,
    const float* __restrict__ w2, const float* __restrict__ b2,
    float* __restrict__ out,   // N x 384 (cols 0..127 written here)
    float* __restrict__ y1,    // N x 128
    float* __restrict__ y2,    // N x 128
    int n_rows)
{
  const int lane = threadIdx.x & 31;
  const int wave = threadIdx.x >> 5;     // 0..7 -> column tile
  const int half = lane >> 4;            // 0 or 1
  const int lm   = lane & 15;

  const int m0 = blockIdx.x * 16;
  const int n0 = wave * 16;
  const int nn = n0 + lm;                // this lane's output column

  int m = m0 + lm;                       // this lane's A row
  if (m >= n_rows) m = n_rows - 1;       // duplicate read on tail (stores guarded)

  const float* ax = x + (size_t)m * C_DIM + 2 * half;

  const float bias0 = b0[nn];
  const float bias1 = b1[nn];
  const float bias2 = b2[nn];

  // C/D layout: lane l, vgpr j -> row = m0 + 8*half + j, col = nn.
  // Bias depends only on column -> same value in all 8 accumulator regs.
  v8f c0, c1, c2;
  #pragma unroll
  for (int j = 0; j < 8; ++j) { c0[j] = bias0; c1[j] = bias1; c2[j] = bias2; }

  #pragma unroll 4
  for (int k0 = 0; k0 < C_DIM; k0 += 4) {
    // A 16x4: lanes 0-15 hold K = k0+0/k0+1, lanes 16-31 hold K = k0+2/k0+3
    v2f a = *(const v2f*)(ax + k0);
    const int kb = k0 + 2 * half;        // B 4x16: K = vgpr + 2*half
    v2f bb0, bb1, bb2;
    bb0.x = w0[(size_t)kb * O_DIM + nn];
    bb0.y = w0[(size_t)(kb + 1) * O_DIM + nn];
    bb1.x = w1[(size_t)kb * O_DIM + nn];
    bb1.y = w1[(size_t)(kb + 1) * O_DIM + nn];
    bb2.x = w2[(size_t)kb * O_DIM + nn];
    bb2.y = w2[(size_t)(kb + 1) * O_DIM + nn];
    // 8 args: (neg_a, A, neg_b, B, c_mod, C, reuse_a, reuse_b)
    c0 = __builtin_amdgcn_wmma_f32_16x16x4_f32(false, a, false, bb0, (short)0, c0, false, false);
    c1 = __builtin_amdgcn_wmma_f32_16x16x4_f32(false, a, false, bb1, (short)0, c1, false, false);
    c2 = __builtin_amdgcn_wmma_f32_16x16x4_f32(false, a, false, bb2, (short)0, c2, false, false);
  }

  const int r0 = m0 + 8 * half;
  #pragma unroll
  for (int j = 0; j < 8; ++j) {
    const int r = r0 + j;
    if (r < n_rows) {
      out[(size_t)r * OUT_STRIDE + nn] = c0[j];
      y1[(size_t)r * O_DIM + nn]       = c1[j];
      y2[(size_t)r * O_DIM + nn]       = c2[j];
    }
  }
}

// ---------------------------------------------------------------------------
// SpMM scatter: out[row[e], :] += ew[e] * v[col[e], :]
// One edge per wave; each lane handles 4 contiguous columns (128/32).
// Gathers are 512B rows that live in the 192MB L2; scatter via f32 atomics.
// ---------------------------------------------------------------------------
__global__ __launch_bounds__(256) void mixhop_spmm(
    const float* __restrict__ v,     // N x 128
    const float* __restrict__ ew,    // E
    const int*   __restrict__ row,   // E
    const int*   __restrict__ col,   // E
    float* __restrict__ out,         // destination base, row stride = out_stride
    int out_stride,
    int n_edges)
{
  const int lane = threadIdx.x & 31;
  int wid = (int)((blockIdx.x * blockDim.x + threadIdx.x) >> 5);
  const int nwaves = (int)((gridDim.x * blockDim.x) >> 5);
  for (int e = wid; e < n_edges; e += nwaves) {
    const float w = ew[e];
    const int   r = row[e];
    const int   c = col[e];
    const float4 s = ((const float4*)(v + (size_t)c * O_DIM))[lane];
    float* dst = out + (size_t)r * out_stride + lane * 4;
    atomicAdd(dst + 0, w * s.x);
    atomicAdd(dst + 1, w * s.y);
    atomicAdd(dst + 2, w * s.z);
    atomicAdd(dst + 3, w * s.w);
  }
}

// ---------------------------------------------------------------------------
// Launch
// ---------------------------------------------------------------------------
extern "C" void kernel_launch(void* const* d_in, const int* in_sizes, int n_in,
                              void* d_out, int out_size, void* d_ws, size_t ws_size,
                              hipStream_t stream) {
  (void)n_in; (void)out_size; (void)ws_size;
  const float* x  = (const float*)d_in[0];
  const float* ew = (const float*)d_in[1];
  const float* w0 = (const float*)d_in[2];
  const float* b0 = (const float*)d_in[3];
  const float* w1 = (const float*)d_in[4];
  const float* b1 = (const float*)d_in[5];
  const float* w2 = (const float*)d_in[6];
  const float* b2 = (const float*)d_in[7];
  const int* row  = (const int*)d_in[8];
  const int* col  = (const int*)d_in[9];
  float* out = (float*)d_out;
  float* ws  = (float*)d_ws;

  const int N = in_sizes[0] / C_DIM;   // 50000
  const int E = in_sizes[1];           // 600000

  float* y1 = ws;                         // N x 128
  float* y2 = ws + (size_t)N * O_DIM;     // N x 128
  float* t  = ws;                         // aliases y1 (dead after spmm #1)

  // 1) zero the whole output (cols 128..383 are atomic accumulation targets;
  //    cols 0..127 are fully overwritten by the GEMM kernel right after)
  mixhop_zero<<<8192, 256, 0, stream>>>(out, (long)N * OUT_STRIDE);

  // 2) fused triple GEMM with fp32 WMMA
  mixhop_gemm3<<<(N + 15) / 16, 256, 0, stream>>>(
      x, w0, b0, w1, b1, w2, b2, out, y1, y2, N);

  // 3) hop-1: out[:,128:256] += spmm(y1)
  mixhop_spmm<<<4096, 256, 0, stream>>>(y1, ew, row, col, out + 128, OUT_STRIDE, E);

  // 4) zero t (reuses y1's space)
  mixhop_zero<<<4096, 256, 0, stream>>>(t, (long)N * O_DIM);

  // 5) hop-2 first SpMM: t = spmm(y2)
  mixhop_spmm<<<4096, 256, 0, stream>>>(y2, ew, row, col, t, O_DIM, E);

  // 6) hop-2 second SpMM: out[:,256:384] += spmm(t)
  mixhop_spmm<<<4096, 256, 0, stream>>>(t, ew, row, col, out + 256, OUT_STRIDE, E);
}